// AllZeroDigitalFilter_80058190397786
// MI455X (gfx1250) — compile-verified
//
#include <hip/hip_runtime.h>

// AllZeroDigitalFilter on MI455X (gfx1250), fp32 WMMA path.
//
// out[b, n*P+p] = (1-f)*(x window . h[b,n,:]) + f*(x window . h[b,n2,:]),
//   f = p/P, n2 = min(n+1, N-1), taps indexed as x[t-j]*h[j], j=0..255.
//
// Per (b,n) segment one wave32 computes all 80 outputs in a single 16x16
// accumulator by folding p = r + 16c into (row r, col 2c+s):
//   A[r][kk]     = x[tb + r + 64 - kk]         (shared Toeplitz slice, K=320)
//   B[kk][2c+s]  = h_s[kk + 16c - 64]          (shifted filters)
// The x window is staged TIME-REVERSED (xr[u] = x[tb+79-u]) so the A
// fragment reads ascending LDS addresses: A[m][kk] = xr[kk - m + 15] ->
// one ds_load_2addr_b32 per K-step, no register swaps. Zero padding for
// the shifted filters and the 6 unused columns lives in LDS (hp rows
// flanked by zeros; row 2 all-zero) so the inner loop has NO data-
// dependent control flow and EXEC stays all-ones (WMMA requirement).
//
// 80 x V_WMMA_F32_16X16X4_F32 per wave, 6400 waves (~0.5M WMMA) -- well
// under the ~0.45us floor set by 10.5MB of HBM traffic at 23.3 TB/s.

#define MM    255          // filter order (taps = MM+1 = 256)
#define PP    80           // samples per segment
#define NN    800          // segments
#define BB    8            // batch
#define TT    (NN * PP)    // 64000 samples per batch
#define KTOT  320          // padded K dimension (256 taps + 64 shift headroom)
#define WIN   335          // x window: [tb-255, tb+79]
#define HPW   384          // padded filter row width: index kk+16c+1 <= 383

typedef __attribute__((ext_vector_type(2))) float v2f;
typedef __attribute__((ext_vector_type(8))) float v8f;

__global__ __launch_bounds__(32)
void zerodf_wmma_kernel(const float* __restrict__ x,
                        const float* __restrict__ h,
                        float* __restrict__ out)
{
    __shared__ float xr[WIN + 1];     // reversed x window: xr[u]=x[tb+79-u]
    __shared__ float hp[3][HPW];      // [0]=h[n] pad, [1]=h[n2] pad, [2]=zeros
    __shared__ float db[2][PP];       // unblended correlations d0, d1

    const int n    = blockIdx.x;
    const int b    = blockIdx.y;
    const int lane = threadIdx.x;     // wave32: one wave per block

    const int tb = n * PP;
    const long long bT = (long long)b * TT;
    const int n2 = (n + 1 < NN) ? (n + 1) : (NN - 1);
    const long long h0off = ((long long)b * NN + n ) * (MM + 1);
    const long long h1off = ((long long)b * NN + n2) * (MM + 1);

    // ---- stage: zero-fill padded filter rows, then drop taps at +64 ----
    for (int i = lane; i < 3 * HPW; i += 32)
        (&hp[0][0])[i] = 0.0f;
    // same lane rewrites the same LDS word it zeroed -> in-order, no barrier
    for (int j = lane; j <= MM; j += 32) {   // 256/32: uniform trip count
        hp[0][64 + j] = h[h0off + j];
        hp[1][64 + j] = h[h1off + j];
    }
    // ---- stage x window [tb-255, tb+79], reversed ----
    for (int i = lane; i <= WIN; i += 32) {
        int g = tb + (PP - 1) - i;            // time tb+79-i
        xr[i] = (g >= 0 && g < TT) ? x[bT + g] : 0.0f;
    }
    __syncthreads();

    // ---- per-lane WMMA fragment coordinates (ISA 7.12.2 layouts) ----
    // A 16x4 f32: lanes 0-15 hold {K0,K1}, lanes 16-31 {K2,K3}, M = lane&15
    // B 4x16 f32: lanes 0-15 hold rows {K0,K1}, lanes 16-31 {K2,K3}, N = lane&15
    const int  m     = lane & 15;        // A row / B column index
    const int  half2 = (lane >> 4) * 2;  // K sub-offset of this half-wave
    const int  abase = 15 - m;           // A[m][kk] = xr[kk + abase]
    const int  c     = m >> 1;           // output offset group: p = r + 16c
    const int  s     = m & 1;            // filter select: h[n] vs h[n2]
    const bool colA  = (m < 10);         // 10 of 16 columns carry data

    const float* hrow  = hp[colA ? s : 2];      // inactive cols read zeros
    const int    bbase = colA ? (16 * c) : 0;   // B[kk][col] = hrow[kk+bbase]

    v8f acc = {};
    #pragma unroll 8
    for (int k0 = 0; k0 < KTOT; k0 += 4) {
        const int kk = k0 + half2;
        v2f av, bv;
        av.x = xr[kk + abase];            // A[m][kk]
        av.y = xr[kk + abase + 1];        // A[m][kk+1]
        bv.x = hrow[kk + bbase];          // B[kk][col]
        bv.y = hrow[kk + bbase + 1];      // B[kk+1][col]
        // D = A x B + C   (fp32, 16x16x4)
        acc = __builtin_amdgcn_wmma_f32_16x16x4_f32(
            /*neg_a=*/false, av, /*neg_b=*/false, bv,
            /*c_mod=*/(short)0, acc, /*reuse_a=*/false, /*reuse_b=*/false);
    }

    // ---- scatter D (16x16) into d0[80], d1[80] via LDS ----
    // D layout: lane L holds column N=L&15; VGPR v holds row M = v + 8*(L>=16)
    if (colA) {
        const int rbase = (lane >> 4) * 8;
        #pragma unroll
        for (int v = 0; v < 8; ++v) {
            db[s][rbase + v + 16 * c] = acc[v];   // p = r + 16c
        }
    }
    __syncthreads();

    // ---- blend with interpolation weight and store ----
    for (int p = lane; p < PP; p += 32) {
        float f = (float)p * (1.0f / (float)PP);
        out[bT + tb + p] = (1.0f - f) * db[0][p] + f * db[1][p];
    }
}

extern "C" void kernel_launch(void* const* d_in, const int* in_sizes, int n_in,
                              void* d_out, int out_size, void* d_ws, size_t ws_size,
                              hipStream_t stream)
{
    (void)in_sizes; (void)n_in; (void)out_size; (void)d_ws; (void)ws_size;
    const float* x = (const float*)d_in[0];   // (B, T) fp32
    const float* h = (const float*)d_in[1];   // (B, N, M+1) fp32
    float* out = (float*)d_out;               // (B, T) fp32

    dim3 grid(NN, BB);                        // one wave32 block per (b, n)
    zerodf_wmma_kernel<<<grid, 32, 0, stream>>>(x, h, out);
}